// MambaBB_38646115729914
// MI455X (gfx1250) — compile-verified
//
#include <hip/hip_runtime.h>
#include <hip/hip_bf16.h>
#include <math.h>

// ---------------- problem constants (from reference) ----------------
#define D_MODEL   256
#define D_STATE   128
#define HEADDIM   64
#define D_INNER   512            // EXPAND * D_MODEL
#define NHEADS    8              // D_INNER / HEADDIM
#define CONV_DIM  768            // D_INNER + 2*D_STATE
#define D_IN_PROJ 1288           // 2*D_INNER + 2*D_STATE + NHEADS
#define KCONV     4
#define BATCH     16
#define SEQ       1024
#define CHUNK     64
#define NC        16             // SEQ / CHUNK
#define NTOK      (BATCH*SEQ)    // 16384
#define EPS       1e-5f

typedef __attribute__((ext_vector_type(2))) float v2f;
typedef __attribute__((ext_vector_type(8))) float v8f;
typedef __attribute__((ext_vector_type(2))) int   v2i;
typedef __attribute__((ext_vector_type(4))) int   v4i;

// D = A(16x4) * B(4x16) + C, f32, wave32.
__device__ __forceinline__ v8f wmma4(v2f a, v2f b, v8f c) {
  return __builtin_amdgcn_wmma_f32_16x16x4_f32(false, a, false, b, (short)0, c,
                                               false, false);
}

__device__ __forceinline__ float silu_f(float x) {
  return x / (1.0f + expf(-x));
}

// ---------------- async global->LDS copy helpers (CDNA5) ----------------
#if defined(__has_builtin)
#if __has_builtin(__builtin_amdgcn_global_load_async_to_lds_b128) && \
    __has_builtin(__builtin_amdgcn_global_load_async_to_lds_b64)
#define HAVE_ASYNC_LDS 1
#endif
#endif
#ifndef HAVE_ASYNC_LDS
#define HAVE_ASYNC_LDS 0
#endif

typedef __attribute__((address_space(1))) v4i* g4p;
typedef __attribute__((address_space(3))) v4i* l4p;
typedef __attribute__((address_space(1))) v2i* g2p;
typedef __attribute__((address_space(3))) v2i* l2p;

__device__ __forceinline__ void async_b128(const float* g, float* l) {
#if HAVE_ASYNC_LDS
  __builtin_amdgcn_global_load_async_to_lds_b128((g4p)g, (l4p)l, 0, 0);
#else
  *(float4*)l = *(const float4*)g;
#endif
}
__device__ __forceinline__ void async_b64(const float* g, float* l) {
#if HAVE_ASYNC_LDS
  __builtin_amdgcn_global_load_async_to_lds_b64((g2p)g, (l2p)l, 0, 0);
#else
  *(v2f*)l = *(const v2f*)g;
#endif
}
__device__ __forceinline__ void wait_async() {
#if defined(__has_builtin) && __has_builtin(__builtin_amdgcn_s_wait_asynccnt)
  __builtin_amdgcn_s_wait_asynccnt(0);
#else
  asm volatile("s_wait_asynccnt 0x0" ::: "memory");
#endif
}

// =====================================================================
// K1/K10: row-major GEMM  C[M,N] = A[M,K] * B[K,N]  (f32 WMMA)
// Block: 16(M) x 128(N) strip, 8 waves x one 16x16 tile.
// K staged in 32-wide double-buffered LDS tiles filled with async copies.
// Requires: M%16==0, K%32==0. N ragged (stores guarded, loads clamped).
// =====================================================================
#define GEMM_KT   32
#define GEMM_NBLK 128
#define A_STRIDE  34   // padded row stride (floats) -> conflict-free ds reads

__global__ void k_gemm_rr(const float* __restrict__ A, const float* __restrict__ B,
                          float* __restrict__ C, int M, int N, int K,
                          int lda, int ldb, int ldc) {
  __shared__ float As[2][16 * A_STRIDE];
  __shared__ float Bs[2][GEMM_KT * GEMM_NBLK];
  const int tid  = threadIdx.x;
  const int lane = tid & 31;
  const int wave = tid >> 5;
  const int lr   = lane & 15;
  const int half = lane >> 4;
  const int tm   = blockIdx.x << 4;
  const int tn0  = blockIdx.y * GEMM_NBLK;
  const int col  = tn0 + (wave << 4) + lr;
  const bool cok = (col < N);

  // staging assignments (256 threads)
  const int arow = tid >> 4;            // 0..15
  const int acol = (tid & 15) * 2;      // 0,2,..,30
  const int brow = tid >> 3;            // 0..31
  const int bcol = (tid & 7) * 16;      // 0,16,..,112
  const float* Ag = A + (size_t)(tm + arow) * lda + acol;

  const int nst = K / GEMM_KT;
  auto stage = [&](int buf, int kb) {
    async_b64(Ag + kb, &As[buf][arow * A_STRIDE + acol]);
    const float* Bg = B + (size_t)(kb + brow) * ldb;
    float* Bl = &Bs[buf][brow * GEMM_NBLK + bcol];
#pragma unroll
    for (int j = 0; j < 4; ++j) {
      int c = tn0 + bcol + 4 * j;
      if (c > N - 4) c = N - 4;          // clamp: ragged-N, garbage unstored
      async_b128(Bg + c, Bl + 4 * j);
    }
  };

  stage(0, 0);
  wait_async();
  __syncthreads();

  v8f acc = {0.f,0.f,0.f,0.f,0.f,0.f,0.f,0.f};
  for (int s = 0; s < nst; ++s) {
    const int cur = s & 1;
    if (s + 1 < nst) stage(cur ^ 1, (s + 1) * GEMM_KT);   // prefetch next stage
    const float* Asb = As[cur];
    const float* Bsb = Bs[cur];
#pragma unroll
    for (int k = 0; k < GEMM_KT; k += 4) {
      const int k0 = k + 2 * half;
      v2f a = *(const v2f*)(Asb + lr * A_STRIDE + k0);
      v2f b;
      b.x = Bsb[k0 * GEMM_NBLK + (wave << 4) + lr];
      b.y = Bsb[(k0 + 1) * GEMM_NBLK + (wave << 4) + lr];
      acc = wmma4(a, b, acc);
    }
    wait_async();
    __syncthreads();
  }
  if (cok) {
    float* Cr = C + (size_t)(tm + 8 * half) * ldc + col;
#pragma unroll
    for (int v = 0; v < 8; ++v) Cr[(size_t)v * ldc] = acc[v];
  }
}

// =====================================================================
// K2: depthwise causal conv(KCONV) + bias + SiLU over xBC channels
// =====================================================================
__global__ void k_conv_silu(const float* __restrict__ zxbcdt,
                            const float* __restrict__ conv_w,
                            const float* __restrict__ conv_b,
                            float* __restrict__ xBCc) {
  int idx = blockIdx.x * blockDim.x + threadIdx.x;
  if (idx >= NTOK * CONV_DIM) return;
  int ch = idx % CONV_DIM;
  int bt = idx / CONV_DIM;
  int t  = bt % SEQ;
  int b  = bt / SEQ;
  float s = conv_b[ch];
#pragma unroll
  for (int k = 0; k < KCONV; ++k) {
    int tt = t - (KCONV - 1) + k;
    float xv = (tt >= 0)
        ? zxbcdt[(size_t)(b * SEQ + tt) * D_IN_PROJ + D_INNER + ch] : 0.0f;
    s += xv * conv_w[ch * KCONV + k];
  }
  xBCc[(size_t)bt * CONV_DIM + ch] = silu_f(s);
}

// =====================================================================
// K3: dt = softplus(dt_raw + dt_bias)
// =====================================================================
__global__ void k_dt_softplus(const float* __restrict__ zxbcdt,
                              const float* __restrict__ dt_bias,
                              float* __restrict__ dtsp) {
  int idx = blockIdx.x * blockDim.x + threadIdx.x;
  if (idx >= NTOK * NHEADS) return;
  int h = idx % NHEADS;
  int bt = idx / NHEADS;
  float x = zxbcdt[(size_t)bt * D_IN_PROJ + (D_INNER + CONV_DIM) + h] + dt_bias[h];
  dtsp[idx] = (x > 20.0f) ? x : log1pf(expf(x));
}

// =====================================================================
// K4: per-chunk cumulative sum of dt*A  -> Acs (b,c,h,l)
// =====================================================================
__global__ void k_chunk_cumsum(const float* __restrict__ dtsp,
                               const float* __restrict__ A_log,
                               float* __restrict__ Acs) {
  int idx = blockIdx.x * blockDim.x + threadIdx.x;
  if (idx >= BATCH * NC * NHEADS) return;
  int h = idx % NHEADS;
  int bc = idx / NHEADS;
  int c = bc % NC;
  int b = bc / NC;
  float Aneg = -expf(A_log[h]);
  const float* d = dtsp + (size_t)(b * SEQ + c * CHUNK) * NHEADS + h;
  float* out = Acs + (size_t)idx * CHUNK;
  float s = 0.0f;
  for (int l = 0; l < CHUNK; ++l) { s += d[l * NHEADS] * Aneg; out[l] = s; }
}

// =====================================================================
// K5: G[b,c,l,s] = sum_n Cc[l,n] * Bc[s,n]   (64x64, K=128)  WMMA
// =====================================================================
__global__ void k_chunk_G(const float* __restrict__ xBCc, float* __restrict__ G) {
  int bc = blockIdx.x;
  int b = bc / NC, c = bc % NC;
  int lane = threadIdx.x & 31, wave = threadIdx.x >> 5;
  int lr = lane & 15, half = lane >> 4;
  const float* Cb = xBCc + (size_t)(b * SEQ + c * CHUNK) * CONV_DIM + D_INNER + D_STATE;
  const float* Bb = xBCc + (size_t)(b * SEQ + c * CHUNK) * CONV_DIM + D_INNER;
  float* Gout = G + (size_t)bc * CHUNK * CHUNK;
  for (int t = wave; t < 16; t += 8) {
    int tm = (t >> 2) << 4, tn = (t & 3) << 4;
    v8f acc = {0.f,0.f,0.f,0.f,0.f,0.f,0.f,0.f};
    const float* Ap = Cb + (size_t)(tm + lr) * CONV_DIM + 2 * half;
    const float* Bp = Bb + (size_t)(tn + lr) * CONV_DIM + 2 * half; // Bc[s][k], k contiguous
#pragma unroll 4
    for (int kb = 0; kb < D_STATE; kb += 4) {
      v2f a  = *(const v2f*)(Ap + kb);
      v2f bb = *(const v2f*)(Bp + kb);
      acc = wmma4(a, bb, acc);
    }
    float* Cr = Gout + (size_t)(tm + 8 * half) * CHUNK + tn + lr;
#pragma unroll
    for (int v = 0; v < 8; ++v) Cr[v * CHUNK] = acc[v];
  }
}

// =====================================================================
// K6: per (b,c,h):  Y_diag = (G .* L) @ Xd,   states = Xd^T @ (Bc .* decay)
//     y = Y_diag + D*x   written; local states written.
// =====================================================================
__global__ void k_ssd_diag(const float* __restrict__ xBCc,
                           const float* __restrict__ dtsp,
                           const float* __restrict__ Acs,
                           const float* __restrict__ G,
                           const float* __restrict__ Dvec,
                           float* __restrict__ y,
                           float* __restrict__ states) {
  __shared__ float Acs_s[CHUNK];
  __shared__ float dec_s[CHUNK];
  __shared__ float Xd_s[CHUNK * HEADDIM];
  __shared__ float Msh[CHUNK * CHUNK];
  int bch = blockIdx.x;
  int h = bch % NHEADS;
  int bc = bch / NHEADS;
  int c = bc % NC, b = bc / NC;
  int tid = threadIdx.x;
  if (tid < CHUNK) Acs_s[tid] = Acs[(size_t)(bc * NHEADS + h) * CHUNK + tid];
  __syncthreads();
  if (tid < CHUNK) dec_s[tid] = expf(Acs_s[CHUNK - 1] - Acs_s[tid]);
  const float* xbase = xBCc + (size_t)(b * SEQ + c * CHUNK) * CONV_DIM + h * HEADDIM;
  const float* dtb = dtsp + (size_t)(b * SEQ + c * CHUNK) * NHEADS + h;
  for (int i = tid; i < CHUNK * HEADDIM; i += blockDim.x) {
    int l = i >> 6, p = i & 63;
    Xd_s[i] = xbase[(size_t)l * CONV_DIM + p] * dtb[l * NHEADS];
  }
  const float* Gp = G + (size_t)bc * CHUNK * CHUNK;
  for (int i = tid; i < CHUNK * CHUNK; i += blockDim.x) {
    int l = i >> 6, s = i & 63;
    Msh[i] = (s <= l) ? Gp[i] * expf(Acs_s[l] - Acs_s[s]) : 0.0f;
  }
  __syncthreads();
  int lane = tid & 31, wave = tid >> 5, lr = lane & 15, half = lane >> 4;
  float Dh = Dvec[h];
  // ---- Y_diag = M(64x64) @ Xd(64x64) ----
  for (int t = wave; t < 16; t += 8) {
    int tm = (t >> 2) << 4, tn = (t & 3) << 4;
    v8f acc = {0.f,0.f,0.f,0.f,0.f,0.f,0.f,0.f};
    const float* Ap = Msh + (tm + lr) * CHUNK + 2 * half;
    const float* Bp = Xd_s + (2 * half) * HEADDIM + tn + lr;
#pragma unroll 4
    for (int kb = 0; kb < CHUNK; kb += 4) {
      v2f a = *(const v2f*)(Ap + kb);
      v2f bb;
      bb.x = Bp[kb * HEADDIM];
      bb.y = Bp[kb * HEADDIM + HEADDIM];
      acc = wmma4(a, bb, acc);
    }
#pragma unroll
    for (int v = 0; v < 8; ++v) {
      int m = tm + 8 * half + v, n = tn + lr;
      float xv = xbase[(size_t)m * CONV_DIM + n];
      y[(size_t)(b * SEQ + c * CHUNK + m) * D_INNER + h * HEADDIM + n] =
          acc[v] + Dh * xv;
    }
  }
  // ---- states(p,n) = sum_l Xd[l,p] * Bc[l,n] * decay[l] ----
  float* Sout = states + (size_t)(bc * NHEADS + h) * HEADDIM * D_STATE;
  const float* Bb = xBCc + (size_t)(b * SEQ + c * CHUNK) * CONV_DIM + D_INNER;
  for (int t = wave; t < 32; t += 8) {
    int tp = (t >> 3) << 4;       // p tile (4 of)
    int tn = (t & 7) << 4;        // n tile (8 of)
    v8f acc = {0.f,0.f,0.f,0.f,0.f,0.f,0.f,0.f};
#pragma unroll 4
    for (int kb = 0; kb < CHUNK; kb += 4) {
      int k0 = kb + 2 * half;
      v2f a;
      a.x = Xd_s[k0 * HEADDIM + tp + lr];
      a.y = Xd_s[(k0 + 1) * HEADDIM + tp + lr];
      v2f bb;
      bb.x = Bb[(size_t)k0 * CONV_DIM + tn + lr] * dec_s[k0];
      bb.y = Bb[(size_t)(k0 + 1) * CONV_DIM + tn + lr] * dec_s[k0 + 1];
      acc = wmma4(a, bb, acc);
    }
#pragma unroll
    for (int v = 0; v < 8; ++v) {
      int p = tp + 8 * half + v, n = tn + lr;
      Sout[(size_t)p * D_STATE + n] = acc[v];
    }
  }
}

// =====================================================================
// K7: inter-chunk state scan (sequential over 16 chunks), in-place:
//     states[c] := prefix-state entering chunk c
// =====================================================================
__global__ void k_state_scan(const float* __restrict__ Acs,
                             float* __restrict__ states) {
  int bh = blockIdx.x;
  int b = bh / NHEADS, h = bh % NHEADS;
  int tid = threadIdx.x;                 // 256 threads, 32 elems each
  float S[32];
#pragma unroll
  for (int i = 0; i < 32; ++i) S[i] = 0.0f;
  for (int c = 0; c < NC; ++c) {
    size_t idx = (size_t)((b * NC + c) * NHEADS + h);
    float dA = expf(Acs[idx * CHUNK + (CHUNK - 1)]);
    size_t base = idx * (HEADDIM * D_STATE);
#pragma unroll
    for (int i = 0; i < 32; ++i) {
      size_t e = base + tid + i * 256;
      float loc = states[e];
      states[e] = S[i];
      S[i] = dA * S[i] + loc;
    }
  }
}

// =====================================================================
// K8: Y_off = exp(Acs[l]) * (Cc @ prefStates^T);  y += Y_off
// =====================================================================
__global__ void k_ssd_off(const float* __restrict__ xBCc,
                          const float* __restrict__ Acs,
                          const float* __restrict__ states,
                          float* __restrict__ y) {
  __shared__ float eA[CHUNK];
  int bch = blockIdx.x;
  int h = bch % NHEADS;
  int bc = bch / NHEADS;
  int c = bc % NC, b = bc / NC;
  int tid = threadIdx.x;
  if (tid < CHUNK) eA[tid] = expf(Acs[(size_t)(bc * NHEADS + h) * CHUNK + tid]);
  __syncthreads();
  int lane = tid & 31, wave = tid >> 5, lr = lane & 15, half = lane >> 4;
  const float* Cb = xBCc + (size_t)(b * SEQ + c * CHUNK) * CONV_DIM + D_INNER + D_STATE;
  const float* Sp = states + (size_t)(bc * NHEADS + h) * HEADDIM * D_STATE;
  for (int t = wave; t < 16; t += 8) {
    int tm = (t >> 2) << 4, tn = (t & 3) << 4;
    v8f acc = {0.f,0.f,0.f,0.f,0.f,0.f,0.f,0.f};
    const float* Ap = Cb + (size_t)(tm + lr) * CONV_DIM + 2 * half;
    const float* Bp = Sp + (size_t)(tn + lr) * D_STATE + 2 * half; // S[p][n], n contiguous
#pragma unroll 4
    for (int kb = 0; kb < D_STATE; kb += 4) {
      v2f a  = *(const v2f*)(Ap + kb);
      v2f bb = *(const v2f*)(Bp + kb);
      acc = wmma4(a, bb, acc);
    }
#pragma unroll
    for (int v = 0; v < 8; ++v) {
      int m = tm + 8 * half + v, n = tn + lr;
      size_t yi = (size_t)(b * SEQ + c * CHUNK + m) * D_INNER + h * HEADDIM + n;
      y[yi] += eA[m] * acc[v];
    }
  }
}

// =====================================================================
// K9: g = y * silu(z);  g *= rsqrt(mean(g^2)+eps) * norm_w  (in place in y)
// =====================================================================
__global__ void k_gate_rmsnorm(const float* __restrict__ zxbcdt,
                               const float* __restrict__ norm_w,
                               float* __restrict__ y) {
  __shared__ float red[256];
  int bt = blockIdx.x;
  int tid = threadIdx.x;
  const float* z = zxbcdt + (size_t)bt * D_IN_PROJ;
  float* yr = y + (size_t)bt * D_INNER;
  float z0 = z[tid], z1 = z[tid + 256];
  float g0 = yr[tid] * silu_f(z0);
  float g1 = yr[tid + 256] * silu_f(z1);
  red[tid] = g0 * g0 + g1 * g1;
  __syncthreads();
  for (int s = 128; s > 0; s >>= 1) {
    if (tid < s) red[tid] += red[tid + s];
    __syncthreads();
  }
  float scale = rsqrtf(red[0] / (float)D_INNER + EPS);
  yr[tid] = g0 * scale * norm_w[tid];
  yr[tid + 256] = g1 * scale * norm_w[tid + 256];
}

// =====================================================================
// host launcher
// =====================================================================
extern "C" void kernel_launch(void* const* d_in, const int* in_sizes, int n_in,
                              void* d_out, int out_size, void* d_ws, size_t ws_size,
                              hipStream_t stream) {
  const float* u       = (const float*)d_in[0];
  const float* W_in    = (const float*)d_in[1];
  const float* conv_w  = (const float*)d_in[2];
  const float* conv_b  = (const float*)d_in[3];
  const float* dt_bias = (const float*)d_in[4];
  const float* A_log   = (const float*)d_in[5];
  const float* Dvec    = (const float*)d_in[6];
  const float* norm_w  = (const float*)d_in[7];
  const float* W_out   = (const float*)d_in[8];
  float* out = (float*)d_out;
  float* ws = (float*)d_ws;

  // workspace layout (floats)
  size_t OFF_ZX  = 0;                                        // (16384,1288)
  size_t OFF_XBC = OFF_ZX  + (size_t)NTOK * D_IN_PROJ;       // (16384,768)
  size_t OFF_DT  = OFF_XBC + (size_t)NTOK * CONV_DIM;        // (16384,8)
  size_t OFF_ACS = OFF_DT  + (size_t)NTOK * NHEADS;          // (b,c,h,64)
  size_t OFF_G   = OFF_ACS + (size_t)BATCH * NC * NHEADS * CHUNK;      // (b,c,64,64)
  size_t OFF_ST  = OFF_G   + (size_t)BATCH * NC * CHUNK * CHUNK;       // (b,c,h,64,128)
  size_t OFF_Y   = OFF_ST  + (size_t)BATCH * NC * NHEADS * HEADDIM * D_STATE; // (16384,512)

  float* zx   = ws + OFF_ZX;
  float* xBCc = ws + OFF_XBC;
  float* dtsp = ws + OFF_DT;
  float* Acs  = ws + OFF_ACS;
  float* G    = ws + OFF_G;
  float* St   = ws + OFF_ST;
  float* Y    = ws + OFF_Y;

  // 1) in-projection GEMM: (16384 x 256) * (256 x 1288)
  {
    dim3 g(NTOK / 16, (D_IN_PROJ + GEMM_NBLK - 1) / GEMM_NBLK);
    k_gemm_rr<<<g, 256, 0, stream>>>(
        u, W_in, zx, NTOK, D_IN_PROJ, D_MODEL, D_MODEL, D_IN_PROJ, D_IN_PROJ);
  }
  // 2) conv + silu
  {
    int n = NTOK * CONV_DIM;
    k_conv_silu<<<(n + 255) / 256, 256, 0, stream>>>(zx, conv_w, conv_b, xBCc);
  }
  // 3) dt softplus
  {
    int n = NTOK * NHEADS;
    k_dt_softplus<<<(n + 255) / 256, 256, 0, stream>>>(zx, dt_bias, dtsp);
  }
  // 4) chunk cumsum
  {
    int n = BATCH * NC * NHEADS;
    k_chunk_cumsum<<<(n + 255) / 256, 256, 0, stream>>>(dtsp, A_log, Acs);
  }
  // 5) G = C B^T
  k_chunk_G<<<BATCH * NC, 256, 0, stream>>>(xBCc, G);
  // 6) diag + local states
  k_ssd_diag<<<BATCH * NC * NHEADS, 256, 0, stream>>>(xBCc, dtsp, Acs, G, Dvec, Y, St);
  // 7) inter-chunk scan
  k_state_scan<<<BATCH * NHEADS, 256, 0, stream>>>(Acs, St);
  // 8) off-diagonal contribution
  k_ssd_off<<<BATCH * NC * NHEADS, 256, 0, stream>>>(xBCc, Acs, St, Y);
  // 9) gate + rmsnorm (in place in Y)
  k_gate_rmsnorm<<<NTOK, 256, 0, stream>>>(zx, norm_w, Y);
  // 10) out-projection GEMM: (16384 x 512) * (512 x 256)
  {
    dim3 g(NTOK / 16, (D_MODEL + GEMM_NBLK - 1) / GEMM_NBLK);
    k_gemm_rr<<<g, 256, 0, stream>>>(
        Y, W_out, out, NTOK, D_MODEL, D_INNER, D_INNER, D_MODEL, D_MODEL);
  }
}